// PredictModel_6700148981854
// MI455X (gfx1250) — compile-verified
//
#include <hip/hip_runtime.h>
#include <hip/hip_bf16.h>

// MI455X (gfx1250) implementation of SSD-style detection post-processing.
//
// Roofline: confidences = 340 MB dominates; HBM @ 23.3 TB/s -> ~15-20 us floor.
// Structure:
//   K1 decode_boxes      : trivial, 16 MB in / 16 MB out
//   K2 softmax_sums_wmma : one streaming pass over confidences; 81-class
//                          exp-row-sum done as WMMA-with-ones (f32 16x16x4),
//                          16 priors per wave, 20 WMMAs per tile. fp32 kept
//                          because scores sit at the 0.01 threshold.
//   K3 topk_nms          : per (b,c) workgroup; strided column re-reads hit
//                          the 192 MB L2 (one batch = 10.6 MB). Histogram
//                          threshold selection -> 512-wide bitonic sort ->
//                          greedy NMS in LDS -> masked write.

typedef float v2f __attribute__((ext_vector_type(2)));
typedef float v8f __attribute__((ext_vector_type(8)));

constexpr int B = 32;
constexpr int P = 32768;
constexpr int C = 81;
constexpr int K = 200;
constexpr float CONF_T = 0.01f;
constexpr float NMS_T = 0.5f;

// ---------------------------------------------------------------- K1: decode
__global__ void decode_boxes(const float* __restrict__ loc,
                             const float* __restrict__ priors,
                             float4* __restrict__ boxes) {
  int idx = blockIdx.x * blockDim.x + threadIdx.x;  // idx = b*P + p
  if (idx >= B * P) return;
  int p = idx & (P - 1);
  float4 l  = ((const float4*)loc)[idx];
  float4 pr = ((const float4*)priors)[p];
  float cx = pr.x + l.x * 0.1f * pr.z;
  float cy = pr.y + l.y * 0.1f * pr.w;
  float w  = pr.z * __expf(l.z * 0.2f);
  float h  = pr.w * __expf(l.w * 0.2f);
  float4 out;
  out.x = cx - 0.5f * w;
  out.y = cy - 0.5f * h;
  out.z = cx + 0.5f * w;
  out.w = cy + 0.5f * h;
  boxes[idx] = out;
}

// ------------------------------------------- K2: softmax denominators (WMMA)
// One wave -> 16 priors. A(16x4) holds exp(conf) chunks, B(4x16) = ones, so
// D[m,n] = sum_k exp(conf[m,k]) for every n. Accumulate 20 chunks (classes
// 0..79), add class 80 in VALU, store inv_sum per prior.
__global__ void softmax_sums_wmma(const float* __restrict__ conf,
                                  float* __restrict__ invs) {
  int tid  = threadIdx.x;
  int wave = tid >> 5;
  int lid  = tid & 31;
  int gw   = blockIdx.x * (blockDim.x >> 5) + wave;  // global wave index
  int b     = gw / (P / 16);
  int pbase = (gw % (P / 16)) * 16;

  // Lane L carries prior row (L&15); half (L>>4) selects class pair within a
  // K=4 chunk per the 32-bit A-matrix layout. With B==ones the row sum is
  // correct for any K permutation.
  const float* rowbase =
      conf + ((size_t)b * P + pbase + (lid & 15)) * C + 2 * (lid >> 4);

  v2f bones; bones.x = 1.0f; bones.y = 1.0f;
  v8f acc = {};

#pragma unroll 4
  for (int k0 = 0; k0 < 80; k0 += 4) {
    v2f a;
    a.x = __expf(rowbase[k0]);
    a.y = __expf(rowbase[k0 + 1]);
    acc = __builtin_amdgcn_wmma_f32_16x16x4_f32(
        /*neg_a=*/false, a, /*neg_b=*/false, bones,
        /*c_mod=*/(short)0, acc, /*reuse_a=*/false, /*reuse_b=*/false);
  }

  // D layout: VGPR r -> row r (lanes 0-15), row r+8 (lanes 16-31).
  // Writers: lanes 0..7 cover priors 0..7, lanes 16..23 cover priors 8..15.
  int r = lid & 15;
  if (r < 8) {
    int prior = (lid >> 4) * 8 + r;
    float s = acc[0];
    if (r == 1) s = acc[1];
    else if (r == 2) s = acc[2];
    else if (r == 3) s = acc[3];
    else if (r == 4) s = acc[4];
    else if (r == 5) s = acc[5];
    else if (r == 6) s = acc[6];
    else if (r == 7) s = acc[7];
    // remainder class 80
    s += __expf(conf[((size_t)b * P + pbase + prior) * C + 80]);
    invs[(size_t)b * P + pbase + prior] = 1.0f / s;
  }
}

// ------------------------------------------------ K3: top-k + NMS + writeout
__global__ void __launch_bounds__(256)
topk_nms(const float* __restrict__ conf, const float* __restrict__ invs,
         const float4* __restrict__ boxes, float* __restrict__ out) {
  const int c = blockIdx.x;   // class
  const int b = blockIdx.y;   // batch
  const int tid = threadIdx.x;

  __shared__ unsigned hist[512];
  __shared__ float cs[512];
  __shared__ int   ci[512];
  __shared__ float4 kb[K];
  __shared__ int keep[K];
  __shared__ int s_T;
  __shared__ int s_cnt;

  for (int i = tid; i < 512; i += 256) hist[i] = 0u;
  if (tid == 0) { s_cnt = 0; s_T = 0; }
  __syncthreads();

  const float* col = conf + (size_t)b * P * C + c;   // stride C floats
  const float* isr = invs + (size_t)b * P;

  // Pass 1: float-bit histogram of masked scores (64 bins per octave).
  for (int p = tid; p < P; p += 256) {
    __builtin_prefetch(col + (size_t)(p + 2048) * C, 0, 1);
    float s = __expf(col[(size_t)p * C]) * isr[p];
    if (s > CONF_T) {
      unsigned bin = (__float_as_uint(s) >> 17) - 7680u;  // >=17 since s>0.01
      if (bin > 511u) bin = 511u;
      atomicAdd(&hist[bin], 1u);
    }
  }
  __syncthreads();

  // Suffix scan from the top bin: smallest score bin whose cumulative count
  // reaches K. If fewer than K positives exist, T stays 0 (take all).
  if (tid == 0) {
    unsigned acc = 0;
    int t = 0;
    for (int i = 511; i >= 0; --i) {
      acc += hist[i];
      if (acc >= (unsigned)K) { t = i; break; }
    }
    s_T = t;
  }
  __syncthreads();
  const int T = s_T;

  // Pass 2: collect candidates at/above threshold bin (L2-resident re-read).
  for (int p = tid; p < P; p += 256) {
    float s = __expf(col[(size_t)p * C]) * isr[p];
    if (s > CONF_T) {
      unsigned bin = (__float_as_uint(s) >> 17) - 7680u;
      if (bin > 511u) bin = 511u;
      if ((int)bin >= T) {
        int pos = atomicAdd(&s_cnt, 1);
        if (pos < 512) { cs[pos] = s; ci[pos] = p; }
      }
    }
  }
  __syncthreads();

  int cnt = s_cnt; if (cnt > 512) cnt = 512;
  for (int i = tid; i < 512; i += 256)
    if (i >= cnt) { cs[i] = 0.0f; ci[i] = 0; }   // pad: zero score -> masked row
  __syncthreads();

  // Bitonic sort, 512 elems, descending by score, ties by ascending index
  // (matches jax top_k tie order).
  for (int kk = 2; kk <= 512; kk <<= 1) {
    for (int j = kk >> 1; j > 0; j >>= 1) {
      for (int t2 = tid; t2 < 512; t2 += 256) {
        int ixj = t2 ^ j;
        if (ixj > t2) {
          float sa = cs[t2], sb = cs[ixj];
          int ia = ci[t2], ib = ci[ixj];
          bool aWorse = (sa < sb) || (sa == sb && ia > ib);
          bool descRegion = ((t2 & kk) == 0);
          if (aWorse == descRegion) {
            cs[t2] = sb; cs[ixj] = sa;
            ci[t2] = ib; ci[ixj] = ia;
          }
        }
      }
      __syncthreads();
    }
  }

  // Gather top-K boxes.
  for (int i = tid; i < K; i += 256) {
    kb[i] = boxes[(size_t)b * P + ci[i]];
    keep[i] = 1;
  }
  __syncthreads();

  // Greedy NMS: sequential over i, parallel over j>i.
  for (int i = 0; i < K - 1; ++i) {
    if (keep[i]) {
      int j = tid;
      if (j > i && j < K && keep[j]) {
        float4 bi = kb[i], bj = kb[j];
        float aw = fmaxf(bi.z - bi.x, 0.0f), ah = fmaxf(bi.w - bi.y, 0.0f);
        float bw = fmaxf(bj.z - bj.x, 0.0f), bh = fmaxf(bj.w - bj.y, 0.0f);
        float areaI = aw * ah, areaJ = bw * bh;
        float lx = fmaxf(bi.x, bj.x), ly = fmaxf(bi.y, bj.y);
        float rx = fminf(bi.z, bj.z), ry = fminf(bi.w, bj.w);
        float iw = fmaxf(rx - lx, 0.0f), ih = fmaxf(ry - ly, 0.0f);
        float inter = iw * ih;
        float iou = inter / (areaI + areaJ - inter + 1e-9f);
        if (iou > NMS_T) keep[j] = 0;
      }
    }
    __syncthreads();
  }

  // Masked writeout: (B, C, K, 5), background class and sub-threshold zeroed.
  float* orow = out + ((size_t)b * C + c) * K * 5;
  for (int i = tid; i < K; i += 256) {
    float s = cs[i];
    float m = (keep[i] && (s > CONF_T) && (c != 0)) ? 1.0f : 0.0f;
    float4 bx = kb[i];
    orow[i * 5 + 0] = bx.x * m;
    orow[i * 5 + 1] = bx.y * m;
    orow[i * 5 + 2] = bx.z * m;
    orow[i * 5 + 3] = bx.w * m;
    orow[i * 5 + 4] = s * m;
  }
}

// ----------------------------------------------------------------- launcher
extern "C" void kernel_launch(void* const* d_in, const int* in_sizes, int n_in,
                              void* d_out, int out_size, void* d_ws, size_t ws_size,
                              hipStream_t stream) {
  const float* conf   = (const float*)d_in[0];  // (B, P, C)
  const float* loc    = (const float*)d_in[1];  // (B, P, 4)
  const float* priors = (const float*)d_in[2];  // (P, 4)
  float* out = (float*)d_out;                   // (B, C, K, 5)

  // workspace: [invs: B*P floats][boxes: B*P float4]  => 20 MB
  float*  invs  = (float*)d_ws;
  float4* boxes = (float4*)((char*)d_ws + (size_t)B * P * sizeof(float));

  decode_boxes<<<dim3((B * P) / 256), dim3(256), 0, stream>>>(loc, priors, boxes);
  // one wave per 16 priors, 8 waves per block: B*P/16/8 = 8192 blocks
  softmax_sums_wmma<<<dim3((B * P) / (16 * 8)), dim3(256), 0, stream>>>(conf, invs);
  topk_nms<<<dim3(C, B), dim3(256), 0, stream>>>(conf, invs, boxes, out);
}